// RelationalGraphAttentionLayer_30193620091225
// MI455X (gfx1250) — compile-verified
//
#include <hip/hip_runtime.h>
#include <hip/hip_bf16.h>

typedef __attribute__((ext_vector_type(2))) float v2f;
typedef __attribute__((ext_vector_type(8))) float v8f;

#define IN_DIM   512
#define OUT_DIM  512
#define N_REL    4

// GEMM tiling: block = 256 threads = 8 wave32s; block tile 32(M) x 64(N), K-tile 32.
#define BM 32
#define BN 64
#define BK 32
#define AS_STRIDE 34   // padded LDS stride (dwords) -> conflict-spread, keeps 8B align for float2
#define BS_STRIDE 34   // B stored transposed [col][k]

// ---------------- coverage kernels: which (rel, node) segments are non-empty ----------------

__global__ void zero_mark_kernel(unsigned int* __restrict__ mark, int total) {
    int i = blockIdx.x * blockDim.x + threadIdx.x;
    if (i < total) mark[i] = 0u;
}

// Entries 0..E-1: real edges (rel = edge_rel[e], v = edge_index[1][e]).
// Entries E..E+n*R-1: reference self-loops: self_nodes = repeat(arange(n),R) -> i//R,
//                                           self_rel   = repeat(arange(R),n) -> i//n.
__global__ void mark_segments_kernel(const int* __restrict__ edge_index,
                                     const int* __restrict__ edge_rel,
                                     unsigned int* __restrict__ mark,
                                     int n, int E) {
    int i = blockIdx.x * blockDim.x + threadIdx.x;
    int total = E + n * N_REL;
    if (i < E) {
        int v = edge_index[E + i];   // node_out = edge_index[1][i]
        int r = edge_rel[i];
        mark[r * n + v] = 1u;        // idempotent store; races benign
    } else if (i < total) {
        int s = i - E;
        int v = s / N_REL;           // self_nodes[s]
        int r = s / n;               // self_rel[s]
        mark[r * n + v] = 1u;
    }
}

// scale[v] = (#non-empty segments over r) / R
__global__ void count_scale_kernel(const unsigned int* __restrict__ mark,
                                   float* __restrict__ scale, int n) {
    int v = blockIdx.x * blockDim.x + threadIdx.x;
    if (v < n) {
        unsigned int c = mark[v] + mark[n + v] + mark[2 * n + v] + mark[3 * n + v];
        scale[v] = (float)c * (1.0f / (float)N_REL);
    }
}

// ---------------- fused WMMA GEMM + bias + per-row scale + ELU ----------------
// out[v, c] = elu( ((x @ W)[v, c] + Wb[c]) * scale[v] )

__global__ __launch_bounds__(256) void rgat_gemm_elu_kernel(
    const float* __restrict__ x,     // [n, 512]
    const float* __restrict__ W,     // [512, 512]
    const float* __restrict__ Wb,    // [512]
    const float* __restrict__ scale, // [n]
    float* __restrict__ out,         // [n, 512]
    int n)
{
    __shared__ float As[BM * AS_STRIDE];   // [row][k]
    __shared__ float Bs[BN * BS_STRIDE];   // transposed: [col][k]

    const int t     = threadIdx.x;
    const int lane  = t & 31;
    const int wave  = t >> 5;
    const int wm    = wave >> 2;           // 0..1  (16-row subtile)
    const int wn    = wave & 3;            // 0..3  (16-col subtile)
    const int lhalf = lane >> 4;           // 0/1
    const int lmod  = lane & 15;

    const int rowBase = blockIdx.y * BM;
    const int colBase = blockIdx.x * BN;

    v8f acc = {};

    for (int k0 = 0; k0 < IN_DIM; k0 += BK) {
        // ---- stage A tile: 32x32 floats, one float4 per thread (coalesced rows) ----
        {
            int row = t >> 3;
            int cg  = t & 7;
            float4 v = make_float4(0.f, 0.f, 0.f, 0.f);
            int gr = rowBase + row;
            if (gr < n)
                v = *reinterpret_cast<const float4*>(x + (size_t)gr * IN_DIM + k0 + cg * 4);
            float* dst = &As[row * AS_STRIDE + cg * 4];
            dst[0] = v.x; dst[1] = v.y; dst[2] = v.z; dst[3] = v.w;
        }
        // ---- stage B tile: 32(k) x 64(col), written transposed into LDS ----
        #pragma unroll
        for (int j = 0; j < 2; ++j) {
            int idx = t + j * 256;
            int kr  = idx >> 4;
            int cg  = idx & 15;
            float4 v = *reinterpret_cast<const float4*>(
                W + (size_t)(k0 + kr) * OUT_DIM + colBase + cg * 4);
            Bs[(cg * 4 + 0) * BS_STRIDE + kr] = v.x;
            Bs[(cg * 4 + 1) * BS_STRIDE + kr] = v.y;
            Bs[(cg * 4 + 2) * BS_STRIDE + kr] = v.z;
            Bs[(cg * 4 + 3) * BS_STRIDE + kr] = v.w;
        }
        // prefetch next K-tile into cache (global_prefetch_b8)
        if (k0 + BK < IN_DIM) {
            int gr = rowBase + (t >> 3);
            if (gr < n)
                __builtin_prefetch(x + (size_t)gr * IN_DIM + (k0 + BK) + (t & 7) * 4, 0, 1);
            __builtin_prefetch(W + (size_t)(k0 + BK + (t >> 4)) * OUT_DIM + colBase + (t & 15) * 4, 0, 1);
        }
        __syncthreads();

        // ---- 8 x V_WMMA_F32_16X16X4_F32 per K-tile per wave ----
        // A frag (16x4 f32): lane holds A[m][kp], A[m][kp+1]; m=lane%16, kp=(lane/16)*2
        // B frag (4x16 f32): lane holds B[kp][nc], B[kp+1][nc]; nc=lane%16 (Bs is [col][k])
        #pragma unroll
        for (int kk = 0; kk < BK; kk += 4) {
            int kp = kk + lhalf * 2;
            v2f a = *reinterpret_cast<const v2f*>(&As[(wm * 16 + lmod) * AS_STRIDE + kp]);
            v2f b = *reinterpret_cast<const v2f*>(&Bs[(wn * 16 + lmod) * BS_STRIDE + kp]);
            acc = __builtin_amdgcn_wmma_f32_16x16x4_f32(
                /*neg_a=*/false, a, /*neg_b=*/false, b,
                /*c_mod=*/(short)0, acc, /*reuse_a=*/false, /*reuse_b=*/false);
        }
        __syncthreads();
    }

    // ---- epilogue: bias + coverage scale + ELU ----
    // C/D layout: vgpr j, lane L -> row = j + 8*(L/16), col = L%16
    const int col = colBase + wn * 16 + lmod;
    const float bias = Wb[col];
    #pragma unroll
    for (int j = 0; j < 8; ++j) {
        int row = rowBase + wm * 16 + j + 8 * lhalf;
        if (row < n) {
            float v = (acc[j] + bias) * scale[row];
            float r = (v > 0.0f) ? v : expm1f(v);
            out[(size_t)row * OUT_DIM + col] = r;
        }
    }
}

// ---------------- launch ----------------

extern "C" void kernel_launch(void* const* d_in, const int* in_sizes, int n_in,
                              void* d_out, int out_size, void* d_ws, size_t ws_size,
                              hipStream_t stream) {
    const float* x          = (const float*)d_in[0];
    const int*   edge_index = (const int*)  d_in[1];
    const int*   edge_rel   = (const int*)  d_in[2];
    // d_in[3] edge_weight: cancels exactly (softmax weights sum to 1 per segment)
    const float* W          = (const float*)d_in[4];
    const float* Wb         = (const float*)d_in[5];
    // d_in[6] a_w, d_in[7] a_b: cancel exactly for the same reason
    float* out = (float*)d_out;

    const int n = in_sizes[0] / IN_DIM;   // 10000
    const int E = in_sizes[2];            // 320000

    // workspace layout
    unsigned int* mark  = (unsigned int*)d_ws;                       // n*R u32
    float*        scale = (float*)((char*)d_ws + (size_t)n * N_REL * sizeof(unsigned int));

    const int nseg = n * N_REL;
    zero_mark_kernel<<<(nseg + 255) / 256, 256, 0, stream>>>(mark, nseg);

    const int total = E + nseg;
    mark_segments_kernel<<<(total + 255) / 256, 256, 0, stream>>>(edge_index, edge_rel, mark, n, E);

    count_scale_kernel<<<(n + 255) / 256, 256, 0, stream>>>(mark, scale, n);

    dim3 grid(OUT_DIM / BN, (n + BM - 1) / BM);   // (8, 313)
    rgat_gemm_elu_kernel<<<grid, 256, 0, stream>>>(x, W, Wb, scale, out, n);
}